// WARPLoss_6493990551871
// MI455X (gfx1250) — compile-verified
//
#include <hip/hip_runtime.h>

typedef __attribute__((ext_vector_type(2))) float v2f;
typedef __attribute__((ext_vector_type(8))) float v8f;

#define ROWS 16
#define COLS 256
#define CP   (COLS + 4)   // pad: banks differ by 4 per row -> conflict-free WMMA A loads
#define TMAX 255

__device__ __forceinline__ unsigned pcg_hash(unsigned x) {
    x = x * 747796405u + 2891336453u;
    unsigned w = ((x >> ((x >> 28u) + 4u)) ^ x) * 277803737u;
    return (w >> 22u) ^ w;
}

__global__ __launch_bounds__(256) void warp_loss_main(
    const float* __restrict__ inp, const int* __restrict__ tgt,
    float* __restrict__ partials) {
    __shared__ float         s_sgn[ROWS][CP];   // (tgt==0 ? x : -x)
    __shared__ unsigned char s_tgt[ROWS][COLS];
    __shared__ float         s_rw[COLS];        // rank_weights
    __shared__ float         s_S[ROWS];         // per-row S_b
    __shared__ int           s_nneg[ROWS];
    __shared__ float         s_part[8];

    const int tid  = threadIdx.x;
    const int row0 = blockIdx.x * ROWS;

    // gfx1250 prefetch of this block's tile (global_prefetch_b8)
    __builtin_prefetch(&inp[row0 * COLS + tid], 0, 0);
    __builtin_prefetch(&tgt[row0 * COLS + tid], 0, 0);

    // rank_weights[r] = 1 + r + H_r  (r>=1), rank_weights[0] = 1
    {
        float h = 0.0f;
        for (int k = 1; k <= tid; ++k) h += 1.0f / (float)k;
        s_rw[tid] = (tid == 0) ? 1.0f : (1.0f + (float)tid + h);
    }
    if (tid < ROWS) s_nneg[tid] = 0;
    __syncthreads();

    // Load + sign-transform: thread `tid` owns column `tid` of all 16 rows.
    for (int i = 0; i < ROWS; ++i) {
        const int g = (row0 + i) * COLS + tid;
        const float v = inp[g];
        const int   t = tgt[g];
        s_sgn[i][tid] = (t == 0) ? v : -v;
        s_tgt[i][tid] = (unsigned char)(t != 0);
        if (t == 0) atomicAdd(&s_nneg[i], 1);   // only the count is used -> deterministic
    }
    __syncthreads();

    // Row sums S_b on the matrix pipe: wave 0 computes sgn(16x256) x ones via
    // 64 chained V_WMMA_F32_16X16X4_F32 ops (f32-exact reduction).
    if (tid < 32) {
        const int m     = tid & 15;          // A: lanes 0-15 and 16-31 both hold M=0..15
        const int khalf = (tid >> 4) * 2;    // VGPR0 holds K = {0 | 2}, VGPR1 holds K+1
        v2f bones; bones.x = 1.0f; bones.y = 1.0f;  // all-ones B (layout-invariant)
        v8f acc = {};
        for (int kk = 0; kk < COLS; kk += 4) {
            v2f a;
            a.x = s_sgn[m][kk + khalf];
            a.y = s_sgn[m][kk + khalf + 1];
            acc = __builtin_amdgcn_wmma_f32_16x16x4_f32(
                false, a, false, bones, (short)0, acc, false, false);
        }
        // C/D layout: lane l (N=l&15), VGPR r -> M = r + 8*(l>=16). Column N=0
        // lives in lanes 0 and 16; every column equals the row sum.
        if (m == 0) {
            const int base = (tid >> 4) * 8;
            #pragma unroll
            for (int r = 0; r < 8; ++r) s_S[base + r] = 256.0f + acc[r];
        }
    }
    __syncthreads();

    // Rejection-sampling phase: thread handles (row i, col tid) for i=0..15.
    float local = 0.0f;
    for (int i = 0; i < ROWS; ++i) {
        if (!s_tgt[i][tid]) continue;        // only positives contribute
        const int nneg = s_nneg[i];
        if (nneg == 0) continue;             // degenerate row: no negatives
        const float sc = s_sgn[i][tid];      // = -inp[b,c] for a positive
        unsigned ctr = (unsigned)((row0 + i) * COLS + tid) * 2654435761u;
        int nt = TMAX;
        #pragma unroll 1
        for (int t = 1; t <= TMAX; ++t) {
            // deterministic draw of a uniformly-random NEGATIVE column
            int j = -1;
            #pragma unroll 1
            for (int att = 0; att < 64; ++att) {
                ctr += 0x9E3779B9u;
                const int cand = (int)(pcg_hash(ctr) & (COLS - 1));
                if (!s_tgt[i][cand]) { j = cand; break; }
            }
            if (j < 0) {  // pathological row: deterministic wrap-scan fallback
                const int start = (int)(pcg_hash(ctr + 1u) & (COLS - 1));
                for (int d = 0; d < COLS; ++d) {
                    const int cand = (start + d) & (COLS - 1);
                    if (!s_tgt[i][cand]) { j = cand; break; }
                }
            }
            // margin = inp[j] - inp[c] = sgn[j] + sc
            if (s_sgn[i][j] + sc >= 0.0f) { nt = t; break; }
        }
        const int r = TMAX / nt;             // floor div, r in [1, 255]
        local += s_rw[r] * s_S[i];           // L * S_b folded directly into loss
    }

    // Deterministic block reduction: shuffle tree per wave, fixed-order wave sum.
    #pragma unroll
    for (int off = 16; off > 0; off >>= 1)
        local += __shfl_down(local, off, 32);
    if ((tid & 31) == 0) s_part[tid >> 5] = local;
    __syncthreads();
    if (tid == 0) {
        float s = 0.0f;
        #pragma unroll
        for (int w = 0; w < 8; ++w) s += s_part[w];
        partials[blockIdx.x] = s;
    }
}

__global__ void warp_loss_finalize(const float* __restrict__ partials,
                                   float* __restrict__ out, int n) {
    if (threadIdx.x == 0 && blockIdx.x == 0) {
        float s = 0.0f;
        for (int i = 0; i < n; ++i) s += partials[i];  // fixed order -> deterministic
        out[0] = s;
    }
}

extern "C" void kernel_launch(void* const* d_in, const int* in_sizes, int n_in,
                              void* d_out, int out_size, void* d_ws, size_t ws_size,
                              hipStream_t stream) {
    const float* inp = (const float*)d_in[0];
    const int*   tgt = (const int*)d_in[1];
    float* out      = (float*)d_out;
    float* partials = (float*)d_ws;

    const int B = in_sizes[0] / COLS;   // 1024
    const int nBlocks = B / ROWS;       // 64

    warp_loss_main<<<nBlocks, 256, 0, stream>>>(inp, tgt, partials);
    warp_loss_finalize<<<1, 32, 0, stream>>>(partials, out, nBlocks);
}